// GumbelSinkhorn_17386027614575
// MI455X (gfx1250) — compile-verified
//
#include <hip/hip_runtime.h>
#include <stdint.h>

// Gumbel-Sinkhorn, fully fused, L2-resident iteration with CDNA5 async
// global->LDS double-buffered row staging.
// B=128 matrices of 512x512 fp32. One 1024-thread block (32 wave32 waves)
// per matrix. Wave w owns rows [16w, 16w+16); lane l owns cols [16l, 16l+16).

#define ITERS 60
#define EPSK  1e-20f
#define NB    128
#define NDIM  512

__device__ __forceinline__ float rcpf(float x) { return __builtin_amdgcn_rcpf(x); }

// wave32 xor-butterfly via ds_swizzle_b32 (immediate mask, no address VGPRs)
template <int MASK>
__device__ __forceinline__ float xsh(float v) {
  return __int_as_float(
      __builtin_amdgcn_ds_swizzle(__float_as_int(v), (MASK << 10) | 0x1f));
}
__device__ __forceinline__ float wsum(float v) {
  v += xsh<16>(v); v += xsh<8>(v); v += xsh<4>(v); v += xsh<2>(v); v += xsh<1>(v);
  return v;
}
__device__ __forceinline__ float wmax(float v) {
  v = fmaxf(v, xsh<16>(v)); v = fmaxf(v, xsh<8>(v)); v = fmaxf(v, xsh<4>(v));
  v = fmaxf(v, xsh<2>(v));  v = fmaxf(v, xsh<1>(v));
  return v;
}

__device__ __forceinline__ void ld16(const float* p, float v[16]) {
  const float4* p4 = (const float4*)p;
#pragma unroll
  for (int q = 0; q < 4; ++q) {
    float4 t = p4[q];
    v[4*q+0] = t.x; v[4*q+1] = t.y; v[4*q+2] = t.z; v[4*q+3] = t.w;
  }
}
__device__ __forceinline__ void st16(float* p, const float v[16]) {
  float4* p4 = (float4*)p;
#pragma unroll
  for (int q = 0; q < 4; ++q) {
    float4 t;
    t.x = v[4*q+0]; t.y = v[4*q+1]; t.z = v[4*q+2]; t.w = v[4*q+3];
    p4[q] = t;
  }
}

// Async copy of this lane's 64B row chunk: global (saddr base + 32b voffset)
// -> LDS (per-lane LDS byte address). IOFFSET is added to BOTH addresses
// (CDNA5 ISA), so one lds/voff pair covers all four b128s.
__device__ __forceinline__ void async_row(uint32_t ldsOff, uint32_t gOff,
                                          const float* base) {
  asm volatile(
      "global_load_async_to_lds_b128 %0, %1, %2 offset:0\n\t"
      "global_load_async_to_lds_b128 %0, %1, %2 offset:16\n\t"
      "global_load_async_to_lds_b128 %0, %1, %2 offset:32\n\t"
      "global_load_async_to_lds_b128 %0, %1, %2 offset:48"
      :
      : "v"(ldsOff), "v"(gOff), "s"(base)
      : "memory");
}
#define WAIT_ASYNC(N) asm volatile("s_wait_asynccnt %0" ::"n"(N) : "memory")

__global__ __launch_bounds__(1024) void gumbel_sinkhorn_fused(
    const float* __restrict__ logits, const float* __restrict__ uin,
    float* __restrict__ P) {
  // 128 KB staging: stage[w][buf][col]. stage[w][0] doubles as the per-wave
  // colsum-partial buffer between the last row of an iteration and the first
  // async issue of the next (barrier-protected on both sides).
  __shared__ __align__(16) float stage[32][2][NDIM];
  __shared__ float cs[2][NDIM];  // double-buffered column sums

  const int tid  = threadIdx.x;
  const int wave = tid >> 5;
  const int lane = tid & 31;
  const int j0   = lane << 4;  // 16 cols per lane
  const size_t mat = (size_t)blockIdx.x * (size_t)(NDIM * NDIM);
  float* const pBase = P + mat;  // block-uniform -> SGPR pair for async saddr

  const uint32_t rowB = NDIM * 4u;
  const uint32_t gRow0 = (uint32_t)(((wave * 16) * NDIM + j0) * 4);  // byte off
  const uint32_t lds0 = (uint32_t)(uintptr_t)(&stage[wave][0][j0]);
  const uint32_t lds1 = (uint32_t)(uintptr_t)(&stage[wave][1][j0]);

  float acc[16];
#pragma unroll
  for (int k = 0; k < 16; ++k) acc[k] = 0.f;

  // ---------------- Phase 0: P = softmax(logits + gumbel(u)) per row -------
#pragma unroll 1
  for (int rr = 0; rr < 16; ++rr) {
    const size_t off = mat + (size_t)(wave * 16 + rr) * NDIM + j0;
    float lg[16], uu[16], t[16];
    ld16(logits + off, lg);
    ld16(uin + off, uu);
#pragma unroll
    for (int k = 0; k < 16; ++k)
      t[k] = lg[k] - __logf(-__logf(uu[k] + EPSK) + EPSK);  // logits + gumbel

    float m = t[0];
#pragma unroll
    for (int k = 1; k < 16; ++k) m = fmaxf(m, t[k]);
    m = wmax(m);

    float s = 0.f;
#pragma unroll
    for (int k = 0; k < 16; ++k) { t[k] = __expf(t[k] - m); s += t[k]; }
    s = wsum(s);
    const float is = rcpf(s);
#pragma unroll
    for (int k = 0; k < 16; ++k) { t[k] *= is; acc[k] += t[k]; }
    st16(P + off, t);
  }
  // Deterministic cross-wave colsum reduction (no fp atomics -> replayable).
  st16(&stage[wave][0][j0], acc);
  __syncthreads();
  if (tid < NDIM) {
    float s = 0.f;
#pragma unroll
    for (int w = 0; w < 32; ++w) s += stage[w][0][tid];
    cs[0][tid] = s;
  }
  __syncthreads();

  // ---------------- 60 Sinkhorn iterations, single fused pass each ---------
#pragma unroll 1
  for (int it = 0; it < ITERS; ++it) {
    const int cur = it & 1;

    float ic[16];  // 1/(colsum+eps) for this lane's 16 columns
    {
      float c[16];
      ld16(&cs[cur][j0], c);
#pragma unroll
      for (int k = 0; k < 16; ++k) ic[k] = rcpf(c[k] + EPSK);
    }
#pragma unroll
    for (int k = 0; k < 16; ++k) acc[k] = 0.f;

    async_row(lds0, gRow0, pBase);  // prime pipeline: row 0 -> buf 0

#pragma unroll 1
    for (int rr = 0; rr < 16; ++rr) {
      if (rr < 15) {
        const uint32_t ldsNext = ((rr + 1) & 1) ? lds1 : lds0;
        async_row(ldsNext, gRow0 + (uint32_t)(rr + 1) * rowB, pBase);
        WAIT_ASYNC(4);  // oldest 4 (row rr) retired, in-order
      } else {
        WAIT_ASYNC(0);
      }

      float v[16];
      ld16(&stage[wave][rr & 1][j0], v);
#pragma unroll
      for (int k = 0; k < 16; ++k) v[k] *= ic[k];  // column-normalize

      float rs = 0.f;
#pragma unroll
      for (int k = 0; k < 16; ++k) rs += v[k];
      rs = wsum(rs);
      const float ir = rcpf(rs + EPSK);
#pragma unroll
      for (int k = 0; k < 16; ++k) { v[k] *= ir; acc[k] += v[k]; }  // row-norm + next colsums
      st16(pBase + (size_t)(wave * 16 + rr) * NDIM + j0, v);
    }

    st16(&stage[wave][0][j0], acc);  // union'd partial buffer
    __syncthreads();                 // also releases the global stores
    if (tid < NDIM) {
      float s = 0.f;
#pragma unroll
      for (int w = 0; w < 32; ++w) s += stage[w][0][tid];
      cs[cur ^ 1][tid] = s;
    }
    __syncthreads();
  }
}

extern "C" void kernel_launch(void* const* d_in, const int* in_sizes, int n_in,
                              void* d_out, int out_size, void* d_ws, size_t ws_size,
                              hipStream_t stream) {
  (void)in_sizes; (void)n_in; (void)out_size; (void)d_ws; (void)ws_size;
  const float* logits = (const float*)d_in[0];
  const float* u      = (const float*)d_in[1];
  float* P            = (float*)d_out;
  gumbel_sinkhorn_fused<<<NB, 1024, 0, stream>>>(logits, u, P);
}